// VectorQuantizer_20263655702966
// MI455X (gfx1250) — compile-verified
//
#include <hip/hip_runtime.h>

typedef __attribute__((ext_vector_type(2))) float v2f;
typedef __attribute__((ext_vector_type(8))) float v8f;

#define M_ROWS        131072      // 32*64*64 rows of dim 64
#define KCODES        512
#define DIM           64
#define CHUNK         128         // codes staged per LDS pass
#define LDS_STRIDE    66          // 64 + 2 pad floats (bank-conflict relief)
#define WAVES_PER_BLK 8
#define ROWS_PER_WAVE 16
#define ROWS_PER_BLK  (WAVES_PER_BLK * ROWS_PER_WAVE)   // 128
#define NBLOCKS       (M_ROWS / ROWS_PER_BLK)           // 1024
#define TOTAL_ELEMS   8388608.0f
#define BETA          0.25f

__global__ __launch_bounds__(256)
void vq_main_kernel(const float* __restrict__ z,
                    const float* __restrict__ emb,
                    float* __restrict__ out,
                    float* __restrict__ blk_ssd)
{
    __shared__ float Elds[CHUNK * LDS_STRIDE];   // staged codebook chunk
    __shared__ float enorm[KCODES];              // ||e||^2 per code
    __shared__ float wsum[WAVES_PER_BLK];

    const int tid  = threadIdx.x;
    const int wave = tid >> 5;
    const int lane = tid & 31;
    const int half = lane >> 4;    // 0: lanes 0-15, 1: lanes 16-31
    const int nl   = lane & 15;

    // ---- codebook squared norms (tiny: 512x64) ----
    for (int c = tid; c < KCODES; c += 256) {
        const float* e = emb + (size_t)c * DIM;
        float s = 0.f;
        #pragma unroll 8
        for (int d = 0; d < DIM; ++d) s += e[d] * e[d];
        enorm[c] = s;
    }

    // ---- load A fragments: 16 rows x 64 dims, ISA 16x4 fp32 layout ----
    // lane L holds row (L&15), K pair {k0 + 2*(L>>4), +1} for fragment k0
    const int R    = (blockIdx.x * WAVES_PER_BLK + wave) * ROWS_PER_WAVE;
    const int arow = R + nl;
    const float* zrow = z + (size_t)arow * DIM;

    v2f a[16];
    #pragma unroll
    for (int kk = 0; kk < 16; ++kk) {
        const float* p = zrow + kk * 4 + half * 2;
        v2f t; t[0] = p[0]; t[1] = p[1];
        a[kk] = t;
    }

    float bestd[8];
    int   besti[8];
    #pragma unroll
    for (int i = 0; i < 8; ++i) { bestd[i] = 3.4e38f; besti[i] = 0; }

    // ---- GEMM + running argmin over code tiles (2 tiles in flight) ----
    for (int ch = 0; ch < KCODES / CHUNK; ++ch) {
        __syncthreads();   // protect LDS chunk from previous iteration readers
        for (int idx = tid; idx < CHUNK * DIM; idx += 256) {
            int c = idx >> 6, d = idx & 63;
            Elds[c * LDS_STRIDE + d] = emb[(size_t)(ch * CHUNK + c) * DIM + d];
        }
        __syncthreads();

        for (int t2 = 0; t2 < CHUNK / 32; ++t2) {
            const int t0 = t2 * 2;
            // Preload all B fragments for two 16-code tiles: the 16 DS loads
            // issue back-to-back and are covered by one dscnt wait.
            // B fragment (4x16, K x N): lane L holds col (L&15), K pair {k0+2*(L>>4),+1}
            const float* bb0 = &Elds[((t0    ) * 16 + nl) * LDS_STRIDE + half * 2];
            const float* bb1 = &Elds[((t0 + 1) * 16 + nl) * LDS_STRIDE + half * 2];
            v2f b0[16], b1[16];
            #pragma unroll
            for (int kk = 0; kk < 16; ++kk) {
                v2f x; x[0] = bb0[kk * 4]; x[1] = bb0[kk * 4 + 1]; b0[kk] = x;
                v2f y; y[0] = bb1[kk * 4]; y[1] = bb1[kk * 4 + 1]; b1[kk] = y;
            }

            v8f acc0 = {0.f,0.f,0.f,0.f,0.f,0.f,0.f,0.f};
            v8f acc1 = {0.f,0.f,0.f,0.f,0.f,0.f,0.f,0.f};
            #pragma unroll
            for (int kk = 0; kk < 16; ++kk) {
                acc0 = __builtin_amdgcn_wmma_f32_16x16x4_f32(
                           false, a[kk], false, b0[kk], (short)0, acc0, false, false);
                acc1 = __builtin_amdgcn_wmma_f32_16x16x4_f32(
                           false, a[kk], false, b1[kk], (short)0, acc1, false, false);
            }

            const int   code0 = ch * CHUNK + t0 * 16 + nl;
            const int   code1 = code0 + 16;
            const float en0   = enorm[code0];
            const float en1   = enorm[code1];
            #pragma unroll
            for (int i = 0; i < 8; ++i) {
                float d0 = en0 - 2.0f * acc0[i];   // ||e||^2 - 2 z.e (row term constant)
                if (d0 < bestd[i]) { bestd[i] = d0; besti[i] = code0; }
                float d1 = en1 - 2.0f * acc1[i];
                if (d1 < bestd[i]) { bestd[i] = d1; besti[i] = code1; }
            }
        }
    }

    // ---- cross-lane argmin within each 16-lane half (C/D layout: row per half) ----
    #pragma unroll
    for (int i = 0; i < 8; ++i) {
        float d  = bestd[i];
        int   ix = besti[i];
        #pragma unroll
        for (int off = 8; off >= 1; off >>= 1) {
            float od = __shfl_xor(d,  off, 32);
            int   oi = __shfl_xor(ix, off, 32);
            if (od < d || (od == d && oi < ix)) { d = od; ix = oi; }
        }
        bestd[i] = d; besti[i] = ix;   // all 16 lanes of the half agree now
    }

    // ---- gather, quantized output, indices, loss partials ----
    float* qout = out + 1;
    float* iout = out + 1 + (size_t)M_ROWS * DIM;
    float  ssd  = 0.f;

    #pragma unroll
    for (int i = 0; i < 8; ++i) {
        const int row = R + i + half * 8;       // lanes0-15: R+i, lanes16-31: R+8+i
        const int ix  = besti[i];
        if (nl == 0) iout[row] = (float)ix;

        const float* e  = emb + (size_t)ix * DIM + nl * 4;
        const float* zr = z   + (size_t)row * DIM + nl * 4;
        float*       qp = qout + (size_t)row * DIM + nl * 4;
        #pragma unroll
        for (int d = 0; d < 4; ++d) {
            float zv  = zr[d];
            float ev  = e[d];
            float t   = ev - zv;        // quantized - z_e
            float qst = zv + t;         // straight-through, exact fp order of reference
            qp[d] = qst;
            float df  = qst - zv;
            ssd += df * df;
        }
    }

    // wave reduction (32 lanes)
    #pragma unroll
    for (int off = 16; off >= 1; off >>= 1)
        ssd += __shfl_xor(ssd, off, 32);
    if (lane == 0) wsum[wave] = ssd;
    __syncthreads();
    if (tid == 0) {
        float s = 0.f;
        #pragma unroll
        for (int w = 0; w < WAVES_PER_BLK; ++w) s += wsum[w];
        blk_ssd[blockIdx.x] = s;        // deterministic per-block partial
    }
}

__global__ __launch_bounds__(256)
void vq_finalize_kernel(const float* __restrict__ blk_ssd, float* __restrict__ out)
{
    __shared__ float s[256];
    float acc = 0.f;
    for (int i = threadIdx.x; i < NBLOCKS; i += 256) acc += blk_ssd[i];
    s[threadIdx.x] = acc;
    __syncthreads();
    for (int o = 128; o > 0; o >>= 1) {
        if (threadIdx.x < o) s[threadIdx.x] += s[threadIdx.x + o];
        __syncthreads();
    }
    if (threadIdx.x == 0)
        out[0] = (1.0f + BETA) * s[0] / TOTAL_ELEMS;
}

extern "C" void kernel_launch(void* const* d_in, const int* in_sizes, int n_in,
                              void* d_out, int out_size, void* d_ws, size_t ws_size,
                              hipStream_t stream)
{
    const float* z   = (const float*)d_in[0];   // z_e: 32*64*64*64 fp32
    const float* emb = (const float*)d_in[1];   // embeddings: 512*64 fp32
    float* out = (float*)d_out;                 // [loss | quantized | indices]
    float* blk = (float*)d_ws;                  // NBLOCKS fp32 partials

    vq_main_kernel<<<NBLOCKS, 256, 0, stream>>>(z, emb, out, blk);
    vq_finalize_kernel<<<1, 256, 0, stream>>>(blk, out);
}